// Decoder_72335839199969
// MI455X (gfx1250) — compile-verified
//
#include <hip/hip_runtime.h>
#include <hip/hip_bf16.h>
#include <math.h>

typedef __attribute__((ext_vector_type(2))) float v2f;
typedef __attribute__((ext_vector_type(8))) float v8f;

// ---------------------------------------------------------------------------
// Problem constants
// ---------------------------------------------------------------------------
#define BB  8
#define DD  192
#define LDG 128
#define LPR 1024
#define MTOT (BB * (LDG + LPR))   // 9216 rows for the fused Wa GEMM

// ---------------------------------------------------------------------------
// GEMM 1: O[b,l,e] = sum_d X[b,d,l] * W[e,d] + bias[e]
// A is read "transposed" (conv layout D x L). One wave per 16x16 tile.
// f32 WMMA 16x16x4:
//   A lane layout: m = lane&15, k-pair = (lane>>4)*2  -> {A[m][k], A[m][k+1]}
//   B lane layout: n = lane&15, same k-pair           -> {B[k][n], B[k+1][n]} = {W[n][k], W[n][k+1]}
//   C VGPR i: rows m0+i (lanes 0-15) and m0+i+8 (lanes 16-31)
// ---------------------------------------------------------------------------
__global__ void gemm_xt_kernel(const float* __restrict__ X,   // (B, DD, L)
                               const float* __restrict__ W,   // (DD, DD) row-major
                               const float* __restrict__ bias,
                               float* __restrict__ O,         // (B*L, DD) row-major
                               int L)
{
    const int lane = threadIdx.x;               // 0..31
    const int m0   = blockIdx.x * 16;
    const int n0   = blockIdx.y * 16;
    const int b    = blockIdx.z;

    const float* Xb = X + (size_t)b * DD * L;
    const int ml    = m0 + (lane & 15);
    const int nl    = n0 + (lane & 15);
    const int khalf = (lane >> 4) << 1;         // 0 or 2

    v8f acc = {};
    for (int k0 = 0; k0 < DD; k0 += 4) {
        const int kb = k0 + khalf;
        v2f a;
        a.x = Xb[(size_t)kb * L + ml];
        a.y = Xb[(size_t)(kb + 1) * L + ml];
        v2f bv = *(const v2f*)(W + (size_t)nl * DD + kb);   // W row-major: contiguous pair
        acc = __builtin_amdgcn_wmma_f32_16x16x4_f32(
            false, a, false, bv, (short)0, acc, false, false);
    }

    const float bn    = bias[nl];
    const int   mbase = m0 + ((lane >> 4) << 3);
    for (int i = 0; i < 8; ++i) {
        O[((size_t)(b * L + mbase + i)) * DD + nl] = acc[i] + bn;
    }
}

// ---------------------------------------------------------------------------
// GEMM 2: O[m,n] = sigmoid(sum_k H[m,k] * W[n,k] + bias[n])  (row-major A)
// ---------------------------------------------------------------------------
__global__ void gemm_rm_sigmoid_kernel(const float* __restrict__ H,  // (M, DD)
                                       const float* __restrict__ W,  // (DD, DD)
                                       const float* __restrict__ bias,
                                       float* __restrict__ O)        // (M, DD)
{
    const int lane = threadIdx.x;
    const int m0   = blockIdx.x * 16;
    const int n0   = blockIdx.y * 16;

    const int ml    = m0 + (lane & 15);
    const int nl    = n0 + (lane & 15);
    const int khalf = (lane >> 4) << 1;

    v8f acc = {};
    for (int k0 = 0; k0 < DD; k0 += 4) {
        const int kb = k0 + khalf;
        v2f a  = *(const v2f*)(H + (size_t)ml * DD + kb);
        v2f bv = *(const v2f*)(W + (size_t)nl * DD + kb);
        acc = __builtin_amdgcn_wmma_f32_16x16x4_f32(
            false, a, false, bv, (short)0, acc, false, false);
    }

    const float bn    = bias[nl];
    const int   mbase = m0 + ((lane >> 4) << 3);
    for (int i = 0; i < 8; ++i) {
        const float v = acc[i] + bn;
        O[((size_t)(mbase + i)) * DD + nl] = 1.0f / (1.0f + __expf(-v));
    }
}

// ---------------------------------------------------------------------------
// H1[b,l,d] = (1/Lp) * sum_p relu(d_att[b,l,d] + p_att[b,p,d])
// grid.x = B*LDG blocks, 192 threads (thread = d) -> coalesced row reads
// ---------------------------------------------------------------------------
__global__ void h1_kernel(const float* __restrict__ datt,
                          const float* __restrict__ patt,
                          float* __restrict__ H)
{
    const int row = blockIdx.x;              // b*LDG + l
    const int b   = row >> 7;                // /128
    const int d   = threadIdx.x;

    const float da = datt[(size_t)row * DD + d];
    const float* pa = patt + (size_t)b * LPR * DD + d;

    float acc = 0.0f;
    for (int p = 0; p < LPR; ++p) {
        acc += fmaxf(da + pa[(size_t)p * DD], 0.0f);
    }
    H[(size_t)row * DD + d] = acc * (1.0f / (float)LPR);
}

// ---------------------------------------------------------------------------
// H2[b,p,d] = (1/Ld) * sum_l relu(d_att[b,l,d] + p_att[b,p,d])
// stored at row offset B*LDG inside H
// ---------------------------------------------------------------------------
__global__ void h2_kernel(const float* __restrict__ datt,
                          const float* __restrict__ patt,
                          float* __restrict__ H)
{
    const int row = blockIdx.x;              // b*LPR + p
    const int b   = row >> 10;               // /1024
    const int d   = threadIdx.x;

    const float pav = patt[(size_t)row * DD + d];
    const float* da = datt + (size_t)b * LDG * DD + d;

    float acc = 0.0f;
    for (int l = 0; l < LDG; ++l) {
        acc += fmaxf(pav + da[(size_t)l * DD], 0.0f);
    }
    H[(size_t)(BB * LDG + row) * DD + d] = acc * (1.0f / (float)LDG);
}

// ---------------------------------------------------------------------------
// Gated residual + max-pool over sequence -> pair (B, 2*DD)
// block = 384 threads: t<192 drug channel e=t, else protein channel e=t-192
// ---------------------------------------------------------------------------
__global__ void pool_kernel(const float* __restrict__ drug,   // (B, DD, LDG)
                            const float* __restrict__ prot,   // (B, DD, LPR)
                            const float* __restrict__ att,    // (MTOT, DD) sigmoid outputs
                            float* __restrict__ pair)         // (B, 2*DD)
{
    const int b = blockIdx.x;
    const int t = threadIdx.x;

    if (t < DD) {
        const int e = t;
        const float* x = drug + ((size_t)b * DD + e) * LDG;
        float mx = -INFINITY;
        for (int l = 0; l < LDG; ++l) {
            const float g = 0.5f + att[((size_t)(b * LDG + l)) * DD + e];
            mx = fmaxf(mx, x[l] * g);
        }
        pair[b * (2 * DD) + e] = mx;
    } else {
        const int e = t - DD;
        const float* x = prot + ((size_t)b * DD + e) * LPR;
        float mx = -INFINITY;
        for (int p = 0; p < LPR; ++p) {
            const float g = 0.5f + att[((size_t)(BB * LDG + b * LPR + p)) * DD + e];
            mx = fmaxf(mx, x[p] * g);
        }
        pair[b * (2 * DD) + DD + e] = mx;
    }
}

// ---------------------------------------------------------------------------
// Tiny MLP GEMV: O[m,n] = act(sum_k X[m,k]*W[n,k] + bias[n]); M=8 only
// ---------------------------------------------------------------------------
__global__ void gemv_kernel(const float* __restrict__ X,   // (M, K)
                            const float* __restrict__ W,   // (N, K)
                            const float* __restrict__ bias,
                            float* __restrict__ O,         // (M, N)
                            int M, int N, int K, int leaky)
{
    const int idx = blockIdx.x * blockDim.x + threadIdx.x;
    if (idx >= M * N) return;
    const int m = idx / N;
    const int n = idx % N;

    const float* x = X + (size_t)m * K;
    const float* w = W + (size_t)n * K;
    float acc = 0.0f;
    for (int k = 0; k < K; ++k) acc = fmaf(x[k], w[k], acc);
    acc += bias[n];
    if (leaky) acc = (acc > 0.0f) ? acc : 0.01f * acc;
    O[idx] = acc;
}

// ---------------------------------------------------------------------------
// Launch
// ---------------------------------------------------------------------------
extern "C" void kernel_launch(void* const* d_in, const int* in_sizes, int n_in,
                              void* d_out, int out_size, void* d_ws, size_t ws_size,
                              hipStream_t stream) {
    const float* drug = (const float*)d_in[0];   // (8,192,128)
    const float* prot = (const float*)d_in[1];   // (8,192,1024)
    const float* Wd = (const float*)d_in[2];
    const float* bd = (const float*)d_in[3];
    const float* Wp = (const float*)d_in[4];
    const float* bp = (const float*)d_in[5];
    const float* Wa = (const float*)d_in[6];
    const float* ba = (const float*)d_in[7];
    const float* W1 = (const float*)d_in[8];
    const float* b1 = (const float*)d_in[9];
    const float* W2 = (const float*)d_in[10];
    const float* b2 = (const float*)d_in[11];
    const float* W3 = (const float*)d_in[12];
    const float* b3 = (const float*)d_in[13];
    const float* Wo = (const float*)d_in[14];
    const float* bo = (const float*)d_in[15];
    float* out = (float*)d_out;                  // (8,2)

    // Workspace layout (floats)
    float* ws   = (float*)d_ws;
    const size_t n_datt = (size_t)BB * LDG * DD;          //   196,608
    const size_t n_patt = (size_t)BB * LPR * DD;          // 1,572,864
    const size_t n_H    = (size_t)MTOT * DD;              // 1,769,472
    float* datt = ws;
    float* patt = datt + n_datt;
    float* H    = patt + n_patt;
    float* att  = H + n_H;
    float* pair = att + n_H;                              // 8*384
    float* h1b  = pair + (size_t)BB * 2 * DD;             // 8*1024
    float* h2b  = h1b + (size_t)BB * 1024;                // 8*1024
    float* h3b  = h2b + (size_t)BB * 1024;                // 8*512

    // 1. d_att / p_att projections (f32 WMMA, bias fused)
    gemm_xt_kernel<<<dim3(LDG / 16, DD / 16, BB), 32, 0, stream>>>(drug, Wd, bd, datt, LDG);
    gemm_xt_kernel<<<dim3(LPR / 16, DD / 16, BB), 32, 0, stream>>>(prot, Wp, bp, patt, LPR);

    // 2. ReLU cross-attention means (mean pushed through the linear Wa map)
    h1_kernel<<<BB * LDG, DD, 0, stream>>>(datt, patt, H);
    h2_kernel<<<BB * LPR, DD, 0, stream>>>(datt, patt, H);

    // 3. Fused Wa GEMM + sigmoid over concatenated H1||H2 (9216 x 192 x 192)
    gemm_rm_sigmoid_kernel<<<dim3(MTOT / 16, DD / 16), 32, 0, stream>>>(H, Wa, ba, att);

    // 4. Gated residual + max pool -> pair
    pool_kernel<<<BB, 2 * DD, 0, stream>>>(drug, prot, att, pair);

    // 5. MLP head
    gemv_kernel<<<(BB * 1024 + 255) / 256, 256, 0, stream>>>(pair, W1, b1, h1b, BB, 1024, 2 * DD, 1);
    gemv_kernel<<<(BB * 1024 + 255) / 256, 256, 0, stream>>>(h1b, W2, b2, h2b, BB, 1024, 1024, 1);
    gemv_kernel<<<(BB * 512 + 255) / 256, 256, 0, stream>>>(h2b, W3, b3, h3b, BB, 512, 1024, 1);
    gemv_kernel<<<1, 256, 0, stream>>>(h3b, Wo, bo, out, BB, 2, 512, 0);
}